// HingePairwiseLoss_65532611002859
// MI455X (gfx1250) — compile-verified
//
#include <hip/hip_runtime.h>
#include <hip/hip_bf16.h>

typedef __attribute__((ext_vector_type(2))) float v2f;
typedef __attribute__((ext_vector_type(8))) float v8f;

#define HINGE_MARGIN 0.01f

// Kernel 1: each wave computes a 16-anchor stripe of the pairwise hinge sum.
// For each 16x16 (i,j) tile:  D = A x B + C  with
//   A (16x4 f32) = [ x_i | 1 | 0 | 0 ]      (rank-2 trick)
//   B (4x16 f32) = [ 1 ; -x_j ; 0 ; 0 ]
//   C            = margin broadcast
// => D_ij = margin + x_i - x_j, produced directly in WMMA C/D layout.
// VALU then does relu + (t_j > t_i) rank-mask + accumulate.
// LDS x/t arrays are padded to a multiple of 16 with x=0, t=-inf so the
// inner loop has NO per-lane guards (uniform EXEC, no branches).
__global__ void hinge_pair_wmma_kernel(const float* __restrict__ x,
                                       const float* __restrict__ t,
                                       float* __restrict__ partials,
                                       int N, int Npad) {
    extern __shared__ float smem[];
    float* sx = smem;         // Npad floats
    float* st = smem + Npad;  // Npad floats

    const int tid = threadIdx.x;
    const int nthreads = blockDim.x;
    for (int k = tid; k < Npad; k += nthreads) {
        sx[k] = (k < N) ? x[k] : 0.0f;
        st[k] = (k < N) ? t[k] : -__builtin_inff();  // pad cols never outrank
    }
    __syncthreads();

    const int lane = tid & 31;
    const int wave = tid >> 5;
    const int wavesPerBlock = nthreads >> 5;
    const int ibase = (blockIdx.x * wavesPerBlock + wave) * 16;
    const bool lo = lane < 16;

    // ---- A matrix (loop invariant) ----
    // VGPR0: lanes0-15 = (M=lane, K=0), lanes16-31 = (M, K=2)
    // VGPR1: lanes0-15 = (M, K=1),      lanes16-31 = (M, K=3)
    const int iIdx = ibase + lane;         // meaningful for lanes 0..15
    v2f A;
    A.x = lo ? sx[iIdx] : 0.0f;            // K=0 row: x_i ; K=2 row: 0 (pad x=0)
    A.y = lo ? 1.0f : 0.0f;                // K=1 row: 1   ; K=3 row: 0

    // Per-lane t_i for the 8 D rows this lane holds:
    // D VGPR r: lanes0-15 -> M=r, lanes16-31 -> M=r+8
    const int mOff = lo ? 0 : 8;
    float ti[8];
#pragma unroll
    for (int r = 0; r < 8; ++r) {
        const int ii = ibase + r + mOff;
        ti[r] = (ii < N) ? st[ii] : __builtin_inff();  // pad anchors never lose
    }

    const float bK0 = lo ? 1.0f : 0.0f;    // B K=0 row of ones (K=2 row: 0)

    v8f C;
#pragma unroll
    for (int r = 0; r < 8; ++r) C[r] = HINGE_MARGIN;

    float acc[8];
#pragma unroll
    for (int r = 0; r < 8; ++r) acc[r] = 0.0f;

    const int jlane = lane & 15;           // column N this lane holds in D
#pragma unroll 4
    for (int jbase = 0; jbase < Npad; jbase += 16) {
        const float xj = sx[jbase + jlane];
        const float tj = st[jbase + jlane];

        v2f B;
        B.x = bK0;                 // K=0 row: ones ; K=2 row: 0
        B.y = lo ? -xj : 0.0f;     // K=1 row: -x_j ; K=3 row: 0

        // D = margin + x_i - x_j over the full 16x16 tile
        v8f D = __builtin_amdgcn_wmma_f32_16x16x4_f32(
            false, A, false, B, (short)0, C, false, false);

#pragma unroll
        for (int r = 0; r < 8; ++r) {
            const float h = fmaxf(D[r], 0.0f);          // relu
            acc[r] += (tj > ti[r]) ? h : 0.0f;          // rank mask
        }
    }

    float p = 0.0f;
#pragma unroll
    for (int r = 0; r < 8; ++r) p += acc[r];

    // Deterministic block reduction (reuse LDS after all x/t reads are done)
    __syncthreads();
    smem[tid] = p;
    __syncthreads();
    for (int s = nthreads >> 1; s > 0; s >>= 1) {
        if (tid < s) smem[tid] += smem[tid + s];
        __syncthreads();
    }
    if (tid == 0) partials[blockIdx.x] = smem[0];
}

// Kernel 2: fixed-order final reduce + scale (deterministic, no atomics)
__global__ void hinge_reduce_scale_kernel(const float* __restrict__ partials,
                                          int nPartials,
                                          float* __restrict__ out,
                                          float scale) {
    __shared__ float red[256];
    const int tid = threadIdx.x;
    float v = 0.0f;
    for (int k = tid; k < nPartials; k += 256) v += partials[k];
    red[tid] = v;
    __syncthreads();
    for (int s = 128; s > 0; s >>= 1) {
        if (tid < s) red[tid] += red[tid + s];
        __syncthreads();
    }
    if (tid == 0) out[0] = red[0] * scale;
}

extern "C" void kernel_launch(void* const* d_in, const int* in_sizes, int n_in,
                              void* d_out, int out_size, void* d_ws, size_t ws_size,
                              hipStream_t stream) {
    const float* x = (const float*)d_in[0];   // input
    const float* t = (const float*)d_in[1];   // target
    float* out = (float*)d_out;
    const int N = in_sizes[0];
    const int Npad = (N + 15) & ~15;

    const int threads = 256;                  // 8 waves per block (wave32)
    const int wavesPerBlock = threads / 32;
    const int nITiles = Npad / 16;
    const int blocks = (nITiles + wavesPerBlock - 1) / wavesPerBlock;

    float* partials = (float*)d_ws;
    const size_t shmem = 2 * (size_t)Npad * sizeof(float);

    hinge_pair_wmma_kernel<<<blocks, threads, shmem, stream>>>(x, t, partials, N, Npad);

    const float scale = 2.0f / ((float)N * (float)N);
    hinge_reduce_scale_kernel<<<1, 256, 0, stream>>>(partials, blocks, out, scale);
}